// FISM_48515950576168
// MI455X (gfx1250) — compile-verified
//
#include <hip/hip_runtime.h>
#include <hip/hip_bf16.h>

// ---------------------------------------------------------------------------
// FISM loss on MI455X (gfx1250, wave32).
// GEMMs use exact-fp32 V_WMMA_F32_16X16X4_F32 (precision-matched to the fp32
// reference; the whole working set is
// L2-resident, so the slower f32 WMMA rate costs nothing).
// ---------------------------------------------------------------------------

typedef __attribute__((ext_vector_type(2))) float v2f;
typedef __attribute__((ext_vector_type(8))) float v8f;

#define IN_FEATS 128
#define HID      64
#define NEG      5
#define HALF_BETA  0.005f   // BETA  * 0.5
#define HALF_GAMMA 0.005f   // GAMMA * 0.5

__device__ __forceinline__ float wave_reduce_sum(float v) {
    // wave32 butterfly: every lane ends with the full sum
    #pragma unroll
    for (int off = 16; off > 0; off >>= 1)
        v += __shfl_xor(v, off, 32);
    return v;
}

// ---------------------------------------------------------------------------
// Kernel 1: h = F @ W_enc + b_enc ; P = h @ W_p ; Q = h @ W_q
// One block = 16 rows of the item matrix; 4 waves each own a 16-wide N tile.
// A-frag (16x4 f32): lane L<16 holds (M=L, K={0,1}); lane L>=16 holds (M=L-16,
// K={2,3}) in v2f. B-frag (4x16 f32) mirrors: lanes 0-15 rows K={0,1},
// lanes 16-31 rows K={2,3}, N = lane%16. C/D: VGPR r -> (M = r + 8*(lane/16),
// N = lane%16) per the ISA 16x16 f32 layout.
// ---------------------------------------------------------------------------
__global__ __launch_bounds__(128)
void fism_gemm_kernel(const float* __restrict__ F,  const float* __restrict__ We,
                      const float* __restrict__ be, const float* __restrict__ Wp,
                      const float* __restrict__ Wq, float* __restrict__ P,
                      float* __restrict__ Q,        float* __restrict__ loss,
                      int num_movies)
{
    __shared__ float h[16][HID + 1];   // +1 pad -> conflict-free column reads
    __shared__ float red[4];

    const int wave = threadIdx.x >> 5;   // 0..3
    const int lane = threadIdx.x & 31;
    const int half = lane >> 4;          // selects K pair {0,1} vs {2,3}
    const int lidx = lane & 15;          // M for A-frags, N for B/C/D frags
    const int row0 = blockIdx.x * 16;
    const int n0   = wave * 16;          // this wave's N tile within HID

    // ---- stage 1: h tile (16 x 64), K = 128 -> 32 f32 WMMAs per wave ----
    v8f c = {};
    for (int k0 = 0; k0 < IN_FEATS; k0 += 4) {
        const int ka = k0 + 2 * half;
        const int rr = min(row0 + lidx, num_movies - 1);   // tail clamp
        v2f a, b;
        a.x = F[(size_t)rr * IN_FEATS + ka];
        a.y = F[(size_t)rr * IN_FEATS + ka + 1];
        b.x = We[(size_t)ka * HID + n0 + lidx];
        b.y = We[(size_t)(ka + 1) * HID + n0 + lidx];
        c = __builtin_amdgcn_wmma_f32_16x16x4_f32(false, a, false, b,
                                                  (short)0, c, false, false);
    }
    const float bias = be[n0 + lidx];
    #pragma unroll
    for (int r = 0; r < 8; ++r)
        h[r + 8 * half][n0 + lidx] = c[r] + bias;
    __syncthreads();

    // ---- stage 2: P/Q tiles, K = 64 -> 2 x 16 f32 WMMAs per wave ----
    v8f cp = {}, cq = {};
    for (int k0 = 0; k0 < HID; k0 += 4) {
        const int ka = k0 + 2 * half;
        v2f a, bp, bq;
        a.x  = h[lidx][ka];
        a.y  = h[lidx][ka + 1];
        bp.x = Wp[(size_t)ka * HID + n0 + lidx];
        bp.y = Wp[(size_t)(ka + 1) * HID + n0 + lidx];
        bq.x = Wq[(size_t)ka * HID + n0 + lidx];
        bq.y = Wq[(size_t)(ka + 1) * HID + n0 + lidx];
        cp = __builtin_amdgcn_wmma_f32_16x16x4_f32(false, a, false, bp,
                                                   (short)0, cp, false, false);
        cq = __builtin_amdgcn_wmma_f32_16x16x4_f32(false, a, false, bq,
                                                   (short)0, cq, false, false);
    }

    float reg = 0.0f;
    #pragma unroll
    for (int r = 0; r < 8; ++r) {
        const int m = row0 + r + 8 * half;
        if (m < num_movies) {
            P[(size_t)m * HID + n0 + lidx] = cp[r];
            Q[(size_t)m * HID + n0 + lidx] = cq[r];
            reg += cp[r] * cp[r] + cq[r] * cq[r];
        }
    }
    reg = wave_reduce_sum(reg);
    if (lane == 0) red[wave] = reg;
    __syncthreads();
    if (threadIdx.x == 0) {
        float t = red[0] + red[1] + red[2] + red[3];
        unsafeAtomicAdd(loss, HALF_BETA * t);
    }
}

// ---------------------------------------------------------------------------
// Kernel 2: COO scatter. One wave32 per edge; 2 floats per lane.
// user_emb[u] += v * P[i];  user_deg[u] += v   (hardware f32 atomics in L2).
// ---------------------------------------------------------------------------
__global__ __launch_bounds__(256)
void fism_scatter_kernel(const float* __restrict__ P,   const float* __restrict__ vals,
                         const int* __restrict__ rows,  const int* __restrict__ cols,
                         float* __restrict__ user_emb,  float* __restrict__ user_deg,
                         int nnz)
{
    const int lane = threadIdx.x & 31;
    const int e = (int)((blockIdx.x * blockDim.x + threadIdx.x) >> 5);
    if (e >= nnz) return;

    const int   u = rows[e];
    const int   i = cols[e];
    const float v = vals[e];

    const float2 p = ((const float2*)(P + (size_t)i * HID))[lane];
    if (e + 1 < nnz) {
        // pull next edge's item row toward L2 (global_prefetch_b8)
        __builtin_prefetch(P + (size_t)cols[e + 1] * HID + 2 * lane, 0, 3);
    }

    float* dst = user_emb + (size_t)u * HID + 2 * lane;
    unsafeAtomicAdd(dst,     v * p.x);
    unsafeAtomicAdd(dst + 1, v * p.y);
    if (lane == 0) unsafeAtomicAdd(user_deg + u, v);
}

// ---------------------------------------------------------------------------
// Kernel 3: ratings + squared-hinge loss. One wave32 per positive sample.
// ---------------------------------------------------------------------------
__global__ __launch_bounds__(256)
void fism_rating_kernel(const float* __restrict__ Q,        const float* __restrict__ user_emb,
                        const float* __restrict__ user_deg, const float* __restrict__ b_u,
                        const float* __restrict__ b_i,      const int* __restrict__ rows,
                        const int* __restrict__ cols,       const int* __restrict__ pos_idx,
                        const int* __restrict__ neg_idx,    float* __restrict__ loss,
                        int num_pos)
{
    __shared__ float red[8];
    const int lane = threadIdx.x & 31;
    const int wave = threadIdx.x >> 5;
    const int p = blockIdx.x * 8 + wave;

    float acc = 0.0f;
    if (p < num_pos) {
        const int e  = pos_idx[p];
        const int u  = rows[e];
        const int ip = cols[e];

        const float inv_deg = 1.0f / user_deg[u];
        float2 ue = ((const float2*)(user_emb + (size_t)u * HID))[lane];
        ue.x *= inv_deg; ue.y *= inv_deg;
        const float bu = b_u[u];

        const float2 qp = ((const float2*)(Q + (size_t)ip * HID))[lane];
        const float  d  = wave_reduce_sum(ue.x * qp.x + ue.y * qp.y);
        const float  r_ui = bu + b_i[ip] + d;

        #pragma unroll
        for (int s = 0; s < NEG; ++s) {
            const int j = neg_idx[p * NEG + s];
            const float2 qn = ((const float2*)(Q + (size_t)j * HID))[lane];
            const float  dn = wave_reduce_sum(ue.x * qn.x + ue.y * qn.y);
            const float  r_n  = bu + b_i[j] + dn;
            const float  diff = 1.0f - (r_ui - r_n);
            acc += diff * diff;
        }
    }
    if (lane == 0) red[wave] = acc;
    __syncthreads();
    if (threadIdx.x == 0) {
        float t = 0.0f;
        #pragma unroll
        for (int w = 0; w < 8; ++w) t += red[w];
        unsafeAtomicAdd(loss, 0.5f * t);
    }
}

// ---------------------------------------------------------------------------
// Kernel 4: GAMMA/2 * (|b_u|^2 + |b_i|^2)
// ---------------------------------------------------------------------------
__global__ __launch_bounds__(256)
void fism_biasreg_kernel(const float* __restrict__ b_u, const float* __restrict__ b_i,
                         int nu, int ni, float* __restrict__ loss)
{
    const int stride = (int)(gridDim.x * blockDim.x);
    const int n = nu + ni;
    float v = 0.0f;
    for (int k = (int)(blockIdx.x * blockDim.x + threadIdx.x); k < n; k += stride) {
        const float x = (k < nu) ? b_u[k] : b_i[k - nu];
        v += x * x;
    }
    v = wave_reduce_sum(v);
    if ((threadIdx.x & 31) == 0) unsafeAtomicAdd(loss, HALF_GAMMA * v);
}

// ---------------------------------------------------------------------------
extern "C" void kernel_launch(void* const* d_in, const int* in_sizes, int n_in,
                              void* d_out, int out_size, void* d_ws, size_t ws_size,
                              hipStream_t stream) {
    (void)n_in; (void)out_size; (void)ws_size;

    const float* features = (const float*)d_in[0];
    const float* W_enc    = (const float*)d_in[1];
    const float* b_enc    = (const float*)d_in[2];
    const float* W_p      = (const float*)d_in[3];
    const float* W_q      = (const float*)d_in[4];
    const float* b_u      = (const float*)d_in[5];
    const float* b_i      = (const float*)d_in[6];
    const float* vals     = (const float*)d_in[7];
    const int*   rows     = (const int*)d_in[8];
    const int*   cols     = (const int*)d_in[9];
    const int*   pos_idx  = (const int*)d_in[10];
    const int*   neg_idx  = (const int*)d_in[11];

    const int num_movies = in_sizes[0] / IN_FEATS;   // 20000
    const int num_users  = in_sizes[5];              // 50000
    const int nnz        = in_sizes[7];              // 2,000,000
    const int num_pos    = in_sizes[10];             // 200,000

    // workspace layout: P | Q | user_emb | user_deg   (~23 MB)
    float* P        = (float*)d_ws;
    float* Qm       = P  + (size_t)num_movies * HID;
    float* user_emb = Qm + (size_t)num_movies * HID;
    float* user_deg = user_emb + (size_t)num_users * HID;

    float* loss = (float*)d_out;

    hipMemsetAsync(loss, 0, sizeof(float), stream);
    hipMemsetAsync(user_emb, 0,
                   sizeof(float) * ((size_t)num_users * HID + (size_t)num_users),
                   stream);

    const int gemm_blocks = (num_movies + 15) / 16;
    fism_gemm_kernel<<<gemm_blocks, 128, 0, stream>>>(
        features, W_enc, b_enc, W_p, W_q, P, Qm, loss, num_movies);

    const long long scatter_threads = (long long)nnz * 32;
    fism_scatter_kernel<<<(int)((scatter_threads + 255) / 256), 256, 0, stream>>>(
        P, vals, rows, cols, user_emb, user_deg, nnz);

    fism_rating_kernel<<<(num_pos + 7) / 8, 256, 0, stream>>>(
        Qm, user_emb, user_deg, b_u, b_i, rows, cols, pos_idx, neg_idx,
        loss, num_pos);

    fism_biasreg_kernel<<<256, 256, 0, stream>>>(b_u, b_i, num_users, num_movies, loss);
}